// AutoregressiveHead_52733608460381
// MI455X (gfx1250) — compile-verified
//
#include <hip/hip_runtime.h>

// ---- CDNA5 / gfx1250 fused autoregressive-heads kernel -------------------
// out[b,l] = relu(x[b,:] @ Wx[l] + y[b,parents(l)] @ Wp[l] + b1[l]) @ W2[l] + b2[l]
//
// Roofline: 275 GFLOP of f32 matrix work vs ~100 MB of HBM traffic
// (x 32MB + Wx 64MB; x's 32x reuse across heads lives in the 192MB L2).
// HBM time ~4.3us @ 23.3TB/s dominates f32 WMMA time => stay in full FP32
// (V_WMMA_F32_16X16X4_F32) and optimize the data stream: TDM
// (tensor_load_to_lds) double-buffered global->LDS DMA so the next chunk's
// transfer overlaps the current chunk's WMMA work.

typedef __attribute__((ext_vector_type(2))) float v2f;
typedef __attribute__((ext_vector_type(4))) float f32x4;
typedef __attribute__((ext_vector_type(8))) float v8f;
typedef __attribute__((ext_vector_type(4))) unsigned int u32x4;
typedef __attribute__((ext_vector_type(4))) int i32x4;
typedef __attribute__((ext_vector_type(8))) int i32x8;

#define L_HEADS 32
#define D_DIM   1024
#define H_DIM   512
#define P_PAR   2
#define BM      64               // rows of x per workgroup
#define KC      64               // K (D) chunk staged through LDS
#define NCHUNK  (D_DIM / KC)     // 16

// LDS layout (float units), double-buffered for async TDM pipeline.
// xs row stride 68  (TDM pad: 4 dwords after every 64)  -> conflict-free A reads
// ws row stride 520 (TDM pad: 4 dwords after every 256) -> conflict-free B reads;
//   the mid-row pad at col 256 is a uniform +4 base offset for waves 4..7.
#define XS_STRIDE 68
#define WS_STRIDE 520
#define XS_OFF(b) ((b) * (BM * XS_STRIDE))
#define WS_OFF(b) (2 * BM * XS_STRIDE + (b) * (KC * WS_STRIDE))
#define YS0_OFF   (2 * BM * XS_STRIDE + 2 * KC * WS_STRIDE)
#define YS1_OFF   (YS0_OFF + BM)
#define RS_OFF    (YS1_OFF + BM)
#define LDS_FLOATS (RS_OFF + BM)   // 75456 floats = 301824 B < 320KB/WG

#if defined(__has_builtin)
#if __has_builtin(__builtin_amdgcn_tensor_load_to_lds) && \
    __has_builtin(__builtin_amdgcn_s_wait_tensorcnt)
#define USE_TDM 1
#endif
#endif
#ifndef USE_TDM
#define USE_TDM 0
#endif

#if USE_TDM
// Build a 2-D Tensor-DMA descriptor (CDNA5 ISA) and
// issue TENSOR_LOAD_TO_LDS. data_size=4B. Dim0 = contiguous axis.
__device__ __forceinline__ void tdm_load_2d(unsigned lds_byte_off,
                                            const float* gptr,
                                            unsigned tile_d0, unsigned tile_d1,
                                            unsigned tensor_d0, unsigned tensor_d1,
                                            unsigned stride_d0,
                                            unsigned pad_interval_code,
                                            unsigned pad_amount_code)
{
    unsigned long long ga = (unsigned long long)(const void*)gptr;
    u32x4 g0;
    g0.x = 1u;                                           // count=1 (valid user D#)
    g0.y = lds_byte_off;                                 // LDS byte address
    g0.z = (unsigned)(ga & 0xFFFFFFFFu);                 // global_addr[31:0]
    g0.w = (unsigned)((ga >> 32) & 0x1FFFFFFu)           // global_addr[56:32]
         | (2u << 30);                                   // type=2 ("image")
    i32x8 g1;
    g1[0] = (int)((2u << 16)                             // data_size=4B
                | (1u << 20)                             // pad_enable
                | (pad_interval_code << 22)
                | (pad_amount_code << 25));
    g1[1] = (int)((tensor_d0 & 0xFFFFu) << 16);          // tensor_dim0[15:0]
    g1[2] = (int)(((tensor_d0 >> 16) & 0xFFFFu)          // tensor_dim0[31:16]
                | ((tensor_d1 & 0xFFFFu) << 16));        // tensor_dim1[15:0]
    g1[3] = (int)(((tensor_d1 >> 16) & 0xFFFFu)          // tensor_dim1[31:16]
                | ((tile_d0 & 0xFFFFu) << 16));          // tile_dim0
    g1[4] = (int)(tile_d1 & 0xFFFFu);                    // tile_dim1 (tile_dim2=0)
    g1[5] = (int)stride_d0;                              // tensor_dim0_stride[31:0]
    g1[6] = 0;                                           // stride hi / dim1_stride
    g1[7] = 0;
    i32x4 z4 = (i32x4){0, 0, 0, 0};                      // 2-D: groups 2/3 unused
#if defined(__clang_major__) && (__clang_major__ >= 23)
    i32x8 z8 = (i32x8){0, 0, 0, 0, 0, 0, 0, 0};
    __builtin_amdgcn_tensor_load_to_lds(g0, g1, z4, z4, z8, 0);
#else
    __builtin_amdgcn_tensor_load_to_lds(g0, g1, z4, z4, 0);
#endif
}
#endif

__global__ __launch_bounds__(256)
void AutoregressiveHead_fused_wmma_f32(const float* __restrict__ x,
                                       const float* __restrict__ y,
                                       const float* __restrict__ Wx,
                                       const float* __restrict__ Wp,
                                       const float* __restrict__ b1,
                                       const float* __restrict__ W2,
                                       const float* __restrict__ b2,
                                       const int*   __restrict__ parent_idx,
                                       float* __restrict__ out)
{
    extern __shared__ float lds[];

    const int l    = blockIdx.y;
    const int row0 = blockIdx.x * BM;
    const int tid  = threadIdx.x;
    const int lane = tid & 31;
    const int wave = tid >> 5;

    // Stage per-row parent y values + zero the per-row output accumulator.
    if (tid < BM) {
        const int b  = row0 + tid;
        const int p0 = parent_idx[l * P_PAR + 0];
        const int p1 = parent_idx[l * P_PAR + 1];
        lds[YS0_OFF + tid] = y[(size_t)b * L_HEADS + p0];
        lds[YS1_OFF + tid] = y[(size_t)b * L_HEADS + p1];
        lds[RS_OFF  + tid] = 0.0f;
    }

    // 4x4 grid of 16x16 f32 accumulator tiles per wave (M=64 x N=64 strip).
    v8f acc[4][4];
#pragma unroll
    for (int m = 0; m < 4; ++m)
#pragma unroll
        for (int n = 0; n < 4; ++n)
            acc[m][n] = (v8f){};

    const float* wxl  = Wx + (size_t)l * D_DIM * H_DIM;
    const float* xrow = x + (size_t)row0 * D_DIM;

    const int hcol0 = wave * 64;                        // logical H-column base
    const int wcol0 = hcol0 + (wave >= 4 ? 4 : 0);      // + TDM mid-row pad
    const int cl    = lane & 15;                        // col / row within 16
    const int kg    = (lane >> 4) * 2;                  // A/B lane-group K offset

#if USE_TDM
    // Prologue: kick off chunk 0 DMA into buffer 0 (x tile + Wx tile).
    if (wave == 0) {
        tdm_load_2d(XS_OFF(0) * 4u, xrow,  KC, BM, KC, BM, D_DIM, 5u, 3u);
        tdm_load_2d(WS_OFF(0) * 4u, wxl, H_DIM, KC, H_DIM, KC, H_DIM, 7u, 3u);
    }
#endif

    for (int c = 0; c < NCHUNK; ++c) {
        const int buf = c & 1;

#if USE_TDM
        if (wave == 0) {
            if (c + 1 < NCHUNK) {
                // Issue next chunk into the other buffer (its previous readers
                // finished at the trailing barrier of iteration c-1), then wait
                // until only those 2 new ops are outstanding => chunk c is done
                // (TDM ops from one wave complete in order).
                const int k1 = (c + 1) * KC;
                tdm_load_2d(XS_OFF(buf ^ 1) * 4u, xrow + k1,
                            KC, BM, KC, BM, D_DIM, 5u, 3u);
                tdm_load_2d(WS_OFF(buf ^ 1) * 4u, wxl + (size_t)k1 * H_DIM,
                            H_DIM, KC, H_DIM, KC, H_DIM, 7u, 3u);
                __builtin_amdgcn_s_wait_tensorcnt(2);
            } else {
                __builtin_amdgcn_s_wait_tensorcnt(0);
            }
        }
        __syncthreads();                   // chunk c visible to all waves
#else
        // Fallback staging path (no TDM builtin): cooperative vector loads.
        const int k0f = c * KC;
        __syncthreads();
#pragma unroll
        for (int i = 0; i < 4; ++i) {
            const int idx = tid + i * 256;
            const int r   = idx >> 4;
            const int c4  = idx & 15;
            f32x4 v = *(const f32x4*)(xrow + (size_t)r * D_DIM + k0f + c4 * 4);
            *(f32x4*)(lds + XS_OFF(buf) + r * XS_STRIDE + c4 * 4) = v;
        }
#pragma unroll 8
        for (int i = 0; i < 32; ++i) {
            const int idx = tid + i * 256;
            const int kk  = idx >> 7;
            const int c4  = idx & 127;
            f32x4 v = *(const f32x4*)(wxl + (size_t)(k0f + kk) * H_DIM + c4 * 4);
            *(f32x4*)(lds + WS_OFF(buf) + kk * WS_STRIDE + c4 * 4 +
                      ((c4 >= 64) ? 4 : 0)) = v;
        }
        __syncthreads();
#endif

        const int xsb = XS_OFF(buf);
        const int wsb = WS_OFF(buf);

        // --- compute: 16 K-steps of 4, 16 f32 WMMAs each
#pragma unroll
        for (int kk = 0; kk < KC / 4; ++kk) {
            const int kb = kk * 4 + kg;

            v2f bfrag[4];   // B[kb][col], B[kb+1][col] (K x N, row across lanes)
#pragma unroll
            for (int n = 0; n < 4; ++n) {
                const int col = wcol0 + n * 16 + cl;
                bfrag[n].x = lds[wsb + (kb + 0) * WS_STRIDE + col];
                bfrag[n].y = lds[wsb + (kb + 1) * WS_STRIDE + col];
            }
#pragma unroll
            for (int m = 0; m < 4; ++m) {
                const int row = m * 16 + cl;
                v2f afrag;  // A[row][kb], A[row][kb+1]
                afrag.x = lds[xsb + row * XS_STRIDE + kb + 0];
                afrag.y = lds[xsb + row * XS_STRIDE + kb + 1];
#pragma unroll
                for (int n = 0; n < 4; ++n) {
                    acc[m][n] = __builtin_amdgcn_wmma_f32_16x16x4_f32(
                        /*neg_a=*/false, afrag, /*neg_b=*/false, bfrag[n],
                        /*c_mod=*/(short)0, acc[m][n],
                        /*reuse_a=*/false, /*reuse_b=*/false);
                }
            }
        }
        __syncthreads();                  // chunk c fully consumed before reuse
    }

    // --- fused epilogue: + parent term + b1, ReLU, dot with W2, row-reduce.
    {
        const int rgrp = (lane >> 4) * 8;   // C/D: lanes 16-31 hold M = r+8
        float wp0a[4], wp1a[4], b1a[4], w2a[4];
#pragma unroll
        for (int n = 0; n < 4; ++n) {
            const int col = hcol0 + n * 16 + cl;
            wp0a[n] = Wp[((size_t)l * P_PAR + 0) * H_DIM + col];
            wp1a[n] = Wp[((size_t)l * P_PAR + 1) * H_DIM + col];
            b1a[n]  = b1[(size_t)l * H_DIM + col];
            w2a[n]  = W2[(size_t)l * H_DIM + col];
        }
#pragma unroll
        for (int m = 0; m < 4; ++m) {
            float psum[8];
#pragma unroll
            for (int r = 0; r < 8; ++r) psum[r] = 0.0f;
#pragma unroll
            for (int r = 0; r < 8; ++r) {
                const int row = m * 16 + rgrp + r;
                const float y0 = lds[YS0_OFF + row];
                const float y1 = lds[YS1_OFF + row];
#pragma unroll
                for (int n = 0; n < 4; ++n) {
                    float v = acc[m][n][r] + y0 * wp0a[n] + y1 * wp1a[n] + b1a[n];
                    v = v > 0.0f ? v : 0.0f;
                    psum[r] += v * w2a[n];
                }
            }
#pragma unroll
            for (int r = 0; r < 8; ++r) {
                atomicAdd(&lds[RS_OFF + m * 16 + rgrp + r], psum[r]); // ds_add_f32
            }
        }
    }

    __syncthreads();
    if (tid < BM) {
        out[(size_t)(row0 + tid) * L_HEADS + l] = lds[RS_OFF + tid] + b2[l];
    }
}

extern "C" void kernel_launch(void* const* d_in, const int* in_sizes, int n_in,
                              void* d_out, int out_size, void* d_ws, size_t ws_size,
                              hipStream_t stream) {
    (void)n_in; (void)d_ws; (void)ws_size; (void)out_size;
    const float* x   = (const float*)d_in[0];
    const float* y   = (const float*)d_in[1];
    const float* Wx  = (const float*)d_in[2];
    const float* Wp  = (const float*)d_in[3];
    const float* b1  = (const float*)d_in[4];
    const float* W2  = (const float*)d_in[5];
    const float* b2  = (const float*)d_in[6];
    const int*   pid = (const int*)d_in[7];
    float* out = (float*)d_out;

    const int B = in_sizes[0] / D_DIM;                    // 8192
    const size_t shmem = LDS_FLOATS * sizeof(float);      // ~295 KB (<320 KB/WG)

    // Allow >64KB dynamic LDS (gfx1250 WGP has 320KB). Host-side attr set,
    // not a stream op, so graph capture is unaffected.
    hipFuncSetAttribute((const void*)AutoregressiveHead_fused_wmma_f32,
                        hipFuncAttributeMaxDynamicSharedMemorySize, (int)shmem);

    dim3 grid(B / BM, L_HEADS);                           // (128, 32) workgroups
    dim3 block(256);                                      // 8 wave32 waves
    AutoregressiveHead_fused_wmma_f32<<<grid, block, shmem, stream>>>(
        x, y, Wx, Wp, b1, W2, b2, pid, out);
}